// ChamfersLossDiagonal_6820408066663
// MI455X (gfx1250) — compile-verified
//
#include <hip/hip_runtime.h>

typedef __attribute__((ext_vector_type(2))) float v2f;
typedef __attribute__((ext_vector_type(4))) float v4f;
typedef __attribute__((ext_vector_type(8))) float v8f;

#define SEGS 4  // J-sweep segments per 16-row strip (parallelism multiplier)

// Monotone uint encoding of float: order-preserving, handles negatives.
__device__ __forceinline__ unsigned enc_f32(float f) {
  unsigned u = __float_as_uint(f);
  return ((int)u < 0) ? ~u : (u | 0x80000000u);
}
__device__ __forceinline__ float dec_f32(unsigned u) {
  unsigned b = (u & 0x80000000u) ? (u & 0x7fffffffu) : ~u;
  return __uint_as_float(b);
}

// Prep: pack WMMA operands and init min-buffers in one pass over B*N points.
//   pX[i] = (-2x0, -2x1, -2x2, |x|^2)   (A rows + C diagonal)
//   pY[j] = ( y0,   y1,   y2,  |y|^2)   (B cols with K=3 = |y|^2)
__global__ __launch_bounds__(256) void cd_prep_kernel(
    const float* __restrict__ x, const float* __restrict__ y,
    v4f* __restrict__ px, v4f* __restrict__ py,
    unsigned* __restrict__ dl_enc, unsigned* __restrict__ dr_enc, int total) {
  int i = blockIdx.x * blockDim.x + threadIdx.x;
  if (i >= total) return;
  float x0 = x[i * 3 + 0], x1 = x[i * 3 + 1], x2 = x[i * 3 + 2];
  px[i] = (v4f){-2.0f * x0, -2.0f * x1, -2.0f * x2, x0 * x0 + x1 * x1 + x2 * x2};
  float y0 = y[i * 3 + 0], y1 = y[i * 3 + 1], y2 = y[i * 3 + 2];
  py[i] = (v4f){y0, y1, y2, y0 * y0 + y1 * y1 + y2 * y2};
  dl_enc[i] = 0xFFFFFFFFu;  // greater than any encoded real
  dr_enc[i] = 0xFFFFFFFFu;
}

// Per-tile reduction: update running row-mins, atomically fold this half's
// 8-row column-min into dl (both halves hit the same address; EXEC stays full).
__device__ __forceinline__ void cd_tile_reduce(const v8f& D, v8f& accr,
                                               unsigned* dlp) {
#pragma unroll
  for (int v = 0; v < 8; ++v) accr[v] = fminf(accr[v], D[v]);
  float cm = fminf(fminf(D[0], D[1]), fminf(D[2], D[3]));
  cm = fminf(cm, fminf(fminf(D[4], D[5]), fminf(D[6], D[7])));
  atomicMin(dlp, enc_f32(cm));
}

// One wave per (batch b, 16-row strip I, J-segment). Per tile: one
// v_wmma_f32_16x16x4_f32 yields the full squared-distance tile
// D[m,n] = |x_m|^2 + |y_n|^2 - 2 x_m.y_n  (rx in C, ry in K=3 slot of B).
__global__ __launch_bounds__(256) void chamfer_tiles_kernel(
    const float* __restrict__ px, const float* __restrict__ py,
    unsigned* __restrict__ dl_enc, unsigned* __restrict__ dr_enc,
    int N, int tiles, int B) {
  const int lane = threadIdx.x & 31;
  const int w = (blockIdx.x * blockDim.x + threadIdx.x) >> 5;  // global wave id
  if (w >= B * tiles * SEGS) return;                            // wave-uniform
  const int b = w / (tiles * SEGS);
  const int r = w - b * (tiles * SEGS);
  const int seg = r / tiles;       // block-mates share (b, seg): same B stream
  const int I = r - seg * tiles;
  const int jspan = tiles / SEGS;
  const int j0 = seg * jspan;

  const int half = lane >> 4;   // 0: K=0,1 lanes; 1: K=2,3 lanes
  const int l16 = lane & 15;

  const float* pxb = px + (size_t)b * N * 4;
  const float* pyb = py + (size_t)b * N * 4;

  // A tile: one b64 load; K=3 slot is the constant 1 (select, no branch).
  const int arow = I * 16 + l16;
  v2f A = *(const v2f*)(pxb + (size_t)arow * 4 + half * 2);
  A.y = half ? 1.0f : A.y;

  // C tile: crx[v] = |x_{I*16 + v + 8*half}|^2
  v8f crx;
#pragma unroll
  for (int v = 0; v < 8; ++v) {
    crx[v] = pxb[(size_t)(I * 16 + v + 8 * half) * 4 + 3];
  }

  v8f accr;
#pragma unroll
  for (int v = 0; v < 8; ++v) accr[v] = __builtin_inff();

  // Incrementing pointers: no per-tile 64-bit address rebuild.
  unsigned* dlp = dl_enc + (size_t)b * N + (size_t)(j0 * 16 + l16);
  const float* bptr = pyb + (size_t)(j0 * 16 + l16) * 4 + half * 2;

  // 1-deep software pipeline: next tile's B operand load is in flight
  // while the current tile's WMMA + min tree executes.
  v2f Bm = *(const v2f*)bptr;
  for (int t = 0; t < jspan - 1; ++t) {
    bptr += 64;
    v2f Bn = *(const v2f*)bptr;

    v8f D = __builtin_amdgcn_wmma_f32_16x16x4_f32(
        false, A, false, Bm, (short)0, crx, false, false);
    cd_tile_reduce(D, accr, dlp);
    dlp += 16;
    Bm = Bn;
  }
  {  // last tile
    v8f D = __builtin_amdgcn_wmma_f32_16x16x4_f32(
        false, A, false, Bm, (short)0, crx, false, false);
    cd_tile_reduce(D, accr, dlp);
  }

  // dr: reduce across the 16 lanes of each half (once per wave)
#pragma unroll
  for (int v = 0; v < 8; ++v) {
    float a = accr[v];
    a = fminf(a, __shfl_xor(a, 1, 32));
    a = fminf(a, __shfl_xor(a, 2, 32));
    a = fminf(a, __shfl_xor(a, 4, 32));
    a = fminf(a, __shfl_xor(a, 8, 32));
    accr[v] = a;
  }
  if (l16 == 0) {  // lane 0 -> rows 0..7, lane 16 -> rows 8..15
#pragma unroll
    for (int v = 0; v < 8; ++v) {
      atomicMin(&dr_enc[(size_t)b * N + I * 16 + v + 8 * half], enc_f32(accr[v]));
    }
  }
}

__global__ __launch_bounds__(256) void cd_finalize_kernel(
    const unsigned* __restrict__ dl_enc, const unsigned* __restrict__ dr_enc,
    float* __restrict__ out, int N) {
  __shared__ float s[256];
  const int b = blockIdx.x;
  float acc = 0.0f;
  for (int j = threadIdx.x; j < N; j += 256) {
    acc += dec_f32(dl_enc[(size_t)b * N + j]) + dec_f32(dr_enc[(size_t)b * N + j]);
  }
  s[threadIdx.x] = acc;
  __syncthreads();
  for (int off = 128; off > 0; off >>= 1) {
    if ((int)threadIdx.x < off) s[threadIdx.x] += s[threadIdx.x + off];
    __syncthreads();
  }
  if (threadIdx.x == 0) out[b] = s[0] / (float)N;
}

extern "C" void kernel_launch(void* const* d_in, const int* in_sizes, int n_in,
                              void* d_out, int out_size, void* d_ws, size_t ws_size,
                              hipStream_t stream) {
  const float* x = (const float*)d_in[0];
  const float* y = (const float*)d_in[1];
  float* out = (float*)d_out;

  const int B = out_size;                 // 16
  const int N = in_sizes[0] / (B * 3);    // 2048
  const int tiles = N / 16;               // 128
  const int total = B * N;

  // Workspace layout: [dl_enc u32 BN][dr_enc u32 BN][pX v4f BN][pY v4f BN]
  unsigned* dl_enc = (unsigned*)d_ws;
  unsigned* dr_enc = (unsigned*)((char*)d_ws + (size_t)total * 4);
  v4f* px = (v4f*)((char*)d_ws + (size_t)total * 8);
  v4f* py = (v4f*)((char*)d_ws + (size_t)total * 8 + (size_t)total * 16);

  cd_prep_kernel<<<(total + 255) / 256, 256, 0, stream>>>(
      x, y, px, py, dl_enc, dr_enc, total);

  const int waves = B * tiles * SEGS;          // 8192 waves
  const int blocks = (waves * 32 + 255) / 256; // 8 waves per block
  chamfer_tiles_kernel<<<blocks, 256, 0, stream>>>(
      (const float*)px, (const float*)py, dl_enc, dr_enc, N, tiles, B);

  cd_finalize_kernel<<<B, 256, 0, stream>>>(dl_enc, dr_enc, out, N);
}